// SearchableSEBlock_21955872817184
// MI455X (gfx1250) — compile-verified
//
#include <hip/hip_runtime.h>
#include <math.h>

// Problem constants (match reference)
#define BB   64
#define CC   256
#define HH   64
#define WW   64
#define HID  16
#define CONDN 32
#define RH   64
#define HWN  (HH * WW)            // 4096
#define NY   ((size_t)BB * CC * HWN)  // 67108864 elements of y
#define MIN_CR 0.3f
#define MAX_CR 1.0f
#define BLEND  0.7f

typedef __attribute__((ext_vector_type(2))) float v2f;
typedef __attribute__((ext_vector_type(4))) float v4f;
typedef __attribute__((ext_vector_type(8))) float v8f;

__device__ __forceinline__ v8f wmma_f32_4(v2f a, v2f b, v8f c) {
  // D = A(16x4 f32) x B(4x16 f32) + C(16x16 f32)
  return __builtin_amdgcn_wmma_f32_16x16x4_f32(
      /*neg_a=*/false, a, /*neg_b=*/false, b,
      /*c_mod=*/(short)0, c, /*reuse_a=*/false, /*reuse_b=*/false);
}

__device__ __forceinline__ float sigmoidf_(float x) {
  return 1.0f / (1.0f + __expf(-x));
}

// ---------------------------------------------------------------------------
// Kernel 1: global average pool. One block per (b,c) plane of 4096 floats.
// x is streamed once with NT policy (no L2 reuse possible: 256MB vs 192MB L2).
// ---------------------------------------------------------------------------
__global__ void gap_kernel(const v4f* __restrict__ x4, float* __restrict__ sem) {
  __shared__ float s[256];
  const int t = threadIdx.x;
  const size_t base = (size_t)blockIdx.x * (HWN / 4);  // 1024 float4 per plane
  float acc = 0.0f;
#pragma unroll
  for (int i = 0; i < 4; ++i) {
    v4f v = __builtin_nontemporal_load(&x4[base + t + i * 256]);
    acc += v.x + v.y + v.z + v.w;
  }
  s[t] = acc;
  __syncthreads();
#pragma unroll
  for (int off = 128; off > 0; off >>= 1) {
    if (t < off) s[t] += s[t + off];
    __syncthreads();
  }
  if (t == 0) sem[blockIdx.x] = s[0] * (1.0f / (float)HWN);
}

// ---------------------------------------------------------------------------
// Kernel 2: "head" — cond encoder + selector + rate controller.
// Single block, 256 threads = 8 wave32. All GEMMs via v_wmma_f32_16x16x4_f32.
//
// WMMA f32 16x16x4 per-lane layout (ISA 7.12.2):
//   A (16x4):  a[v], lane L  -> A[L&15][k0 + 2*(L>>4) + v]
//   B (4x16):  b[v], lane L  -> B[k0 + 2*v + (L>>4)][L&15]
//   C (16x16): acc[r], lane L-> C[r + 8*(L>>4)][L&15]
// ---------------------------------------------------------------------------
__global__ void head_kernel(
    const float* __restrict__ sem,       // [64,256] from kernel 1 (ws)
    const float* __restrict__ base_cr,   // [1]
    const float* __restrict__ embed,     // [3,16]
    const float* __restrict__ snr_w1,    // [16,1]
    const float* __restrict__ snr_b1,    // [16]
    const float* __restrict__ snr_w2,    // [16,16]
    const float* __restrict__ snr_b2,    // [16]
    const float* __restrict__ sem_w,     // [16,256]
    const float* __restrict__ cond_w,    // [16,32]
    const float* __restrict__ out_w,     // [256,16]
    const float* __restrict__ r_w1,      // [64,288]
    const float* __restrict__ r_b1,      // [64]
    const float* __restrict__ r_w2,      // [64,64]
    const float* __restrict__ r_b2,      // [64]
    const float* __restrict__ r_w3,      // [1,64]
    const float* __restrict__ r_b3,      // [1]
    const int*   __restrict__ channel_idx,
    const int*   __restrict__ snr_db,
    float* __restrict__ wts,             // [64,256] out (ws)
    int*   __restrict__ kv,              // [64] out (ws)
    float* __restrict__ cr_out)          // [64] out (d_out tail)
{
  __shared__ float s_cond[CONDN];       // cond vector (batch-invariant)
  __shared__ float s_condterm[HID];     // cond @ cond_w.T
  __shared__ float s_fused[BB * HID];   // relu(sem@sem_w.T + condterm)
  __shared__ float s_r1[BB * RH];
  __shared__ float s_r2[BB * RH];

  const int tid  = threadIdx.x;
  const int lane = tid & 31;
  const int wv   = tid >> 5;
  const int lr   = lane & 15;   // low lane id
  const int lh   = lane >> 4;   // lane half (0/1)

  // ---- cond encoder (scalar, identical across batch) ----
  if (tid == 0) {
    const int ci = channel_idx[0];
    const float snr_norm = (float)snr_db[0] / 28.0f;
    float h[HID];
#pragma unroll
    for (int i = 0; i < HID; ++i) {
      float v = snr_norm * snr_w1[i] + snr_b1[i];
      h[i] = v > 0.0f ? v : 0.0f;
    }
#pragma unroll
    for (int j = 0; j < HID; ++j) {
      float v = snr_b2[j];
      for (int i = 0; i < HID; ++i) v += h[i] * snr_w2[j * HID + i];
      s_cond[HID + j] = v > 0.0f ? v : 0.0f;     // snr_vec
      s_cond[j] = embed[ci * HID + j];           // channel_vec
    }
  }
  __syncthreads();
  if (tid < HID) {
    float v = 0.0f;
    for (int j = 0; j < CONDN; ++j) v += s_cond[j] * cond_w[tid * CONDN + j];
    s_condterm[tid] = v;
  }
  __syncthreads();

  // ---- GEMM1: fused = relu(sem[64x256] @ sem_w.T[256x16] + condterm) ----
  if (wv < 4) {
    const int m0 = wv * 16;
    v8f acc = {};
    for (int k0 = 0; k0 < CC; k0 += 4) {
      v2f a, b;
      const int ka = k0 + 2 * lh;
      a.x = sem[(m0 + lr) * CC + ka];
      a.y = sem[(m0 + lr) * CC + ka + 1];
      b.x = sem_w[lr * CC + (k0 + lh)];          // B[k][n] = sem_w[n*256+k]
      b.y = sem_w[lr * CC + (k0 + 2 + lh)];
      acc = wmma_f32_4(a, b, acc);
    }
#pragma unroll
    for (int r = 0; r < 8; ++r) {
      const int m = m0 + r + 8 * lh;
      const float v = acc[r] + s_condterm[lr];
      s_fused[m * HID + lr] = v > 0.0f ? v : 0.0f;
    }
  }
  __syncthreads();

  // ---- GEMM2: weights = sigmoid(fused[64x16] @ out_w.T[16x256]) ----
  for (int t = wv; t < 64; t += 8) {           // 4 M-tiles x 16 N-tiles
    const int m0 = (t & 3) * 16;
    const int n0 = (t >> 2) * 16;
    v8f acc = {};
#pragma unroll
    for (int k0 = 0; k0 < HID; k0 += 4) {
      v2f a, b;
      const int ka = k0 + 2 * lh;
      a.x = s_fused[(m0 + lr) * HID + ka];
      a.y = s_fused[(m0 + lr) * HID + ka + 1];
      b.x = out_w[(n0 + lr) * HID + (k0 + lh)];   // B[k][n] = out_w[n*16+k]
      b.y = out_w[(n0 + lr) * HID + (k0 + 2 + lh)];
      acc = wmma_f32_4(a, b, acc);
    }
#pragma unroll
    for (int r = 0; r < 8; ++r) {
      const int m = m0 + r + 8 * lh;
      wts[m * CC + (n0 + lr)] = sigmoidf_(acc[r]);
    }
  }

  // ---- GEMM3: r1 = relu(rin[64x288] @ r_w1.T[288x64] + b1), rin=[sem|cond] ----
  for (int t = wv; t < 16; t += 8) {           // 4x4 tiles of 16x16
    const int m0 = (t & 3) * 16;
    const int n0 = (t >> 2) * 16;
    v8f acc = {};
    for (int k0 = 0; k0 < CC + CONDN; k0 += 4) {
      v2f a, b;
      const int ka = k0 + 2 * lh;
      const int row = m0 + lr;
      a.x = (ka     < CC) ? sem[row * CC + ka]     : s_cond[ka - CC];
      a.y = (ka + 1 < CC) ? sem[row * CC + ka + 1] : s_cond[ka + 1 - CC];
      b.x = r_w1[(n0 + lr) * (CC + CONDN) + (k0 + lh)];
      b.y = r_w1[(n0 + lr) * (CC + CONDN) + (k0 + 2 + lh)];
      acc = wmma_f32_4(a, b, acc);
    }
#pragma unroll
    for (int r = 0; r < 8; ++r) {
      const int m = m0 + r + 8 * lh;
      const int n = n0 + lr;
      const float v = acc[r] + r_b1[n];
      s_r1[m * RH + n] = v > 0.0f ? v : 0.0f;
    }
  }
  __syncthreads();

  // ---- GEMM4: r2 = relu(r1[64x64] @ r_w2.T[64x64] + b2) ----
  for (int t = wv; t < 16; t += 8) {
    const int m0 = (t & 3) * 16;
    const int n0 = (t >> 2) * 16;
    v8f acc = {};
#pragma unroll
    for (int k0 = 0; k0 < RH; k0 += 4) {
      v2f a, b;
      const int ka = k0 + 2 * lh;
      a.x = s_r1[(m0 + lr) * RH + ka];
      a.y = s_r1[(m0 + lr) * RH + ka + 1];
      b.x = r_w2[(n0 + lr) * RH + (k0 + lh)];
      b.y = r_w2[(n0 + lr) * RH + (k0 + 2 + lh)];
      acc = wmma_f32_4(a, b, acc);
    }
#pragma unroll
    for (int r = 0; r < 8; ++r) {
      const int m = m0 + r + 8 * lh;
      const int n = n0 + lr;
      const float v = acc[r] + r_b2[n];
      s_r2[m * RH + n] = v > 0.0f ? v : 0.0f;
    }
  }
  __syncthreads();

  // ---- rate head: raw -> cr -> k ----
  if (tid < BB) {
    float acc = r_b3[0];
    for (int j = 0; j < RH; ++j) acc += s_r2[tid * RH + j] * r_w3[j];
    const float raw = sigmoidf_(acc);
    const float dyn = MIN_CR + (MAX_CR - MIN_CR) * raw;
    float cr = (1.0f - BLEND) * base_cr[0] + BLEND * dyn;
    cr = fminf(fmaxf(cr, MIN_CR), MAX_CR);
    cr_out[tid] = cr;
    const float cs = fminf(fmaxf(cr, 0.001f), 1.0f);
    int k = (int)roundf(cs * (float)CC);
    k = k < 1 ? 1 : (k > CC ? CC : k);
    kv[tid] = k;
  }
}

// ---------------------------------------------------------------------------
// Kernel 3: per-row rank (stable, matches argsort(argsort(-w))) -> mask.
// One block per batch row, one thread per channel.
// ---------------------------------------------------------------------------
__global__ void mask_kernel(const float* __restrict__ wts,
                            const int* __restrict__ kv,
                            float* __restrict__ msk) {
  __shared__ float row[CC];
  const int b = blockIdx.x;
  const int c = threadIdx.x;
  row[c] = wts[b * CC + c];
  __syncthreads();
  const float w = row[c];
  int rank = 0;
  for (int j = 0; j < CC; ++j) {
    const float wj = row[j];
    rank += (wj > w) || (wj == w && j < c);
  }
  msk[b * CC + c] = (rank < kv[b]) ? 1.0f : 0.0f;
}

// ---------------------------------------------------------------------------
// Kernel 4: y = x * mask. Pure streaming pass: NT loads for x (no reuse
// possible), NT stores for y (write-once, never re-read). Mask stays RT
// (64KB, heavily reused, lives in L2).
// ---------------------------------------------------------------------------
__global__ void scale_kernel(const v4f* __restrict__ x4,
                             const float* __restrict__ msk,
                             v4f* __restrict__ y4) {
  const int idx = blockIdx.x * blockDim.x + threadIdx.x;   // 0..16777215
  const float m = msk[idx >> 10];                          // 1024 float4 / plane
  v4f v = __builtin_nontemporal_load(&x4[idx]);
  v.x *= m; v.y *= m; v.z *= m; v.w *= m;
  __builtin_nontemporal_store(v, &y4[idx]);
}

// ---------------------------------------------------------------------------
extern "C" void kernel_launch(void* const* d_in, const int* in_sizes, int n_in,
                              void* d_out, int out_size, void* d_ws, size_t ws_size,
                              hipStream_t stream) {
  const float* x        = (const float*)d_in[0];
  const float* base_cr  = (const float*)d_in[1];
  const float* embed    = (const float*)d_in[2];
  const float* snr_w1   = (const float*)d_in[3];
  const float* snr_b1   = (const float*)d_in[4];
  const float* snr_w2   = (const float*)d_in[5];
  const float* snr_b2   = (const float*)d_in[6];
  const float* sem_w    = (const float*)d_in[7];
  const float* cond_w   = (const float*)d_in[8];
  const float* out_w    = (const float*)d_in[9];
  const float* r_w1     = (const float*)d_in[10];
  const float* r_b1     = (const float*)d_in[11];
  const float* r_w2     = (const float*)d_in[12];
  const float* r_b2     = (const float*)d_in[13];
  const float* r_w3     = (const float*)d_in[14];
  const float* r_b3     = (const float*)d_in[15];
  const int*   ch_idx   = (const int*)d_in[16];
  const int*   snr_db   = (const int*)d_in[17];

  float* out    = (float*)d_out;
  float* cr_out = out + NY;              // cr_values appended after y

  float* ws    = (float*)d_ws;
  float* sem   = ws;                     // 16384 floats
  float* wts   = ws + 16384;             // 16384 floats
  int*   kv    = (int*)(ws + 32768);     // 64 ints
  float* msk   = ws + 33024;             // 16384 floats

  // 1) GAP: one block per (b,c) plane
  gap_kernel<<<BB * CC, 256, 0, stream>>>((const v4f*)x, sem);

  // 2) head: single block, WMMA GEMM chain
  head_kernel<<<1, 256, 0, stream>>>(
      sem, base_cr, embed, snr_w1, snr_b1, snr_w2, snr_b2,
      sem_w, cond_w, out_w, r_w1, r_b1, r_w2, r_b2, r_w3, r_b3,
      ch_idx, snr_db, wts, kv, cr_out);

  // 3) rank -> mask
  mask_kernel<<<BB, CC, 0, stream>>>(wts, kv, msk);

  // 4) y = x * mask  (67108864 floats = 16777216 float4)
  scale_kernel<<<(int)(NY / 4 / 256), 256, 0, stream>>>(
      (const v4f*)x, msk, (v4f*)out);
}